// SegmentScatteringCNN_74509092651254
// MI455X (gfx1250) — compile-verified
//
#include <hip/hip_runtime.h>

// ---------------------------------------------------------------------------
// SegmentScatteringCNN for MI455X (gfx1250).
// Reformulation: bilinear-upsample + segment-mean + 1x1 conv is linear in
// `features`, so we never materialize the 1 GB upsampled tensor. Instead:
//   A[b,s,k]  : accumulated bilinear weight of src cell k hitting segment s
//               (built with f32 global atomics; 4 MB accumulator, L2-resident)
//   G[b,k,f]  = sum_c feat[b,c,k] * w[f,c]          (f32 WMMA GEMM)
//   cnt[b,s]  = sum_k A[b,s,k]   (bilinear weights sum to 1 per pixel)
//   out[b,s,f]= (sum_k A[b,s,k]*G[b,k,f]) / max(cnt,1) + bias[f]  (f32 WMMA)
// ---------------------------------------------------------------------------

#define NSEG 1024
constexpr int Bn  = 4;
constexpr int Cn  = 256;
constexpr int HF  = 16;
constexpr int WF  = 16;
constexpr int KSP = HF * WF;   // 256 source cells
constexpr int Hn  = 512;
constexpr int Wn  = 512;
constexpr int Fn  = 192;

typedef float v2f __attribute__((ext_vector_type(2)));
typedef float v8f __attribute__((ext_vector_type(8)));

// ---- zero the A accumulator (float4 stores) -------------------------------
__global__ void k_zero_f4(float4* __restrict__ p) {
  int i = blockIdx.x * blockDim.x + threadIdx.x;
  p[i] = make_float4(0.f, 0.f, 0.f, 0.f);
}

// ---- scatter bilinear weights into A[b, seg, cell] ------------------------
__global__ void k_scatter(const int* __restrict__ seg, float* __restrict__ A) {
  int idx = blockIdx.x * blockDim.x + threadIdx.x;       // 0 .. 4*512*512-1
  int b   = idx >> 18;
  int rem = idx & ((1 << 18) - 1);
  int h   = rem >> 9;
  int w   = rem & (Wn - 1);

  // half-pixel bilinear, scale 32: src = (p+0.5)/32 - 0.5
  float sy  = (h + 0.5f) * (1.0f / 32.0f) - 0.5f;
  float fy  = floorf(sy);
  float wy1 = sy - fy, wy0 = 1.0f - wy1;
  int y0 = (int)fy, y1 = y0 + 1;
  y0 = y0 < 0 ? 0 : y0;
  y1 = y1 > HF - 1 ? HF - 1 : y1;

  float sx  = (w + 0.5f) * (1.0f / 32.0f) - 0.5f;
  float fx  = floorf(sx);
  float wx1 = sx - fx, wx0 = 1.0f - wx1;
  int x0 = (int)fx, x1 = x0 + 1;
  x0 = x0 < 0 ? 0 : x0;
  x1 = x1 > WF - 1 ? WF - 1 : x1;

  int s = seg[idx];
  float* base = A + (((size_t)b << 10) + s) * KSP;
  __hip_atomic_fetch_add(base + y0 * WF + x0, wy0 * wx0, __ATOMIC_RELAXED, __HIP_MEMORY_SCOPE_AGENT);
  __hip_atomic_fetch_add(base + y0 * WF + x1, wy0 * wx1, __ATOMIC_RELAXED, __HIP_MEMORY_SCOPE_AGENT);
  __hip_atomic_fetch_add(base + y1 * WF + x0, wy1 * wx0, __ATOMIC_RELAXED, __HIP_MEMORY_SCOPE_AGENT);
  __hip_atomic_fetch_add(base + y1 * WF + x1, wy1 * wx1, __ATOMIC_RELAXED, __HIP_MEMORY_SCOPE_AGENT);
}

// ---- inv_cnt[b,s] = 1 / max(row_sum(A), 1) --------------------------------
__global__ void k_inv_count(const float* __restrict__ A, float* __restrict__ invc) {
  int s = blockIdx.x * blockDim.x + threadIdx.x;         // 0 .. 4095
  const float4* p = (const float4*)(A + (size_t)s * KSP);
  float sum = 0.f;
#pragma unroll 8
  for (int i = 0; i < KSP / 4; ++i) {
    float4 v = p[i];
    sum += v.x + v.y + v.z + v.w;
  }
  invc[s] = 1.0f / fmaxf(sum, 1.0f);
}

// ---- G[b,k,f] = sum_c feat[b,c,k] * w[f,c]   via V_WMMA_F32_16X16X4_F32 ---
// One wave per 16x16 tile of G.  M = KSP(256), N = Fn(192), K = Cn(256).
__global__ void k_gemm_g(const float* __restrict__ feat,
                         const float* __restrict__ w,
                         float* __restrict__ G) {
  int wave = blockIdx.x * (blockDim.x >> 5) + (threadIdx.x >> 5);
  int lane = threadIdx.x & 31;
  int half = lane >> 4;            // K sub-group: {0,1} vs {2,3}
  int l    = lane & 15;
  int b    = wave / ((KSP / 16) * (Fn / 16));          // / 192
  int t    = wave % ((KSP / 16) * (Fn / 16));
  int tm   = (t / (Fn / 16)) << 4;
  int tn   = (t % (Fn / 16)) << 4;

  const float* fb = feat + (size_t)b * Cn * KSP;
  int row  = tm + l;               // A-matrix row (spatial cell)
  int col  = tn + l;               // B-matrix col (output feature)
  int koff = half << 1;

  v8f acc = {};
  for (int k0 = 0; k0 < Cn; k0 += 4) {
    v2f a;                                       // A[m, k0+koff(+1)] = feat[c, m]
    a.x = fb[(k0 + koff)     * KSP + row];
    a.y = fb[(k0 + koff + 1) * KSP + row];
    v2f bb = *(const v2f*)(w + (size_t)col * Cn + k0 + koff);  // B[c, f] = w[f, c]
    acc = __builtin_amdgcn_wmma_f32_16x16x4_f32(false, a, false, bb,
                                                (short)0, acc, false, false);
  }
  float* Gb = G + (size_t)b * KSP * Fn;
#pragma unroll
  for (int r = 0; r < 8; ++r)
    Gb[(tm + r + (half << 3)) * Fn + col] = acc[r];
}

// ---- out[b,s,f] = (A[s,:] . G[:,f]) * inv_cnt[s] + bias[f]   (WMMA) -------
// M = NSEG(1024), N = Fn(192), K = KSP(256).
__global__ void k_gemm_out(const float* __restrict__ A,
                           const float* __restrict__ G,
                           const float* __restrict__ invc,
                           const float* __restrict__ bias,
                           float* __restrict__ out) {
  int wave = blockIdx.x * (blockDim.x >> 5) + (threadIdx.x >> 5);
  int lane = threadIdx.x & 31;
  int half = lane >> 4;
  int l    = lane & 15;
  int b    = wave / ((NSEG / 16) * (Fn / 16));         // / 768
  int t    = wave % ((NSEG / 16) * (Fn / 16));
  int tm   = (t / (Fn / 16)) << 4;
  int tn   = (t % (Fn / 16)) << 4;

  const float* Ab = A + (size_t)b * NSEG * KSP;
  const float* Gb = G + (size_t)b * KSP * Fn;
  int row  = tm + l;
  int col  = tn + l;
  int koff = half << 1;

  v8f acc = {};
  for (int k0 = 0; k0 < KSP; k0 += 4) {
    v2f a = *(const v2f*)(Ab + (size_t)row * KSP + k0 + koff);  // 8B aligned
    v2f bb;
    bb.x = Gb[(k0 + koff)     * Fn + col];
    bb.y = Gb[(k0 + koff + 1) * Fn + col];
    acc = __builtin_amdgcn_wmma_f32_16x16x4_f32(false, a, false, bb,
                                                (short)0, acc, false, false);
  }
#pragma unroll
  for (int r = 0; r < 8; ++r) {
    int rr = tm + r + (half << 3);
    out[(((size_t)b << 10) + rr) * Fn + col] =
        acc[r] * invc[((size_t)b << 10) + rr] + bias[col];
  }
}

// ---- second tuple element: segment ids promoted to float ------------------
__global__ void k_seg_copy(const int* __restrict__ seg, float* __restrict__ dst) {
  int i = blockIdx.x * blockDim.x + threadIdx.x;
  dst[i] = (float)seg[i];
}

extern "C" void kernel_launch(void* const* d_in, const int* in_sizes, int n_in,
                              void* d_out, int out_size, void* d_ws, size_t ws_size,
                              hipStream_t stream) {
  (void)in_sizes; (void)n_in; (void)out_size; (void)ws_size;
  const float* feat = (const float*)d_in[0];   // (4,256,16,16)
  const float* wcv  = (const float*)d_in[1];   // (192,256)
  const float* bias = (const float*)d_in[2];   // (192,)
  const int*   seg  = (const int*)d_in[3];     // (4,512,512)
  float* out = (float*)d_out;

  // workspace layout (floats): A [4*1024*256] | G [4*256*192] | invc [4096]
  float* A    = (float*)d_ws;
  float* G    = A + (size_t)Bn * NSEG * KSP;   // +1,048,576
  float* invc = G + (size_t)Bn * KSP * Fn;     // +196,608

  // 1) zero A (1,048,576 floats = 262,144 float4)
  k_zero_f4<<<1024, 256, 0, stream>>>((float4*)A);
  // 2) scatter bilinear weights (1M pixels)
  k_scatter<<<4096, 256, 0, stream>>>(seg, A);
  // 3) G = feat^T x w^T  (768 waves, 4 waves/block)
  k_gemm_g<<<192, 128, 0, stream>>>(feat, wcv, G);
  // 4) inverse counts from A row sums
  k_inv_count<<<16, 256, 0, stream>>>(A, invc);
  // 5) out = (A x G) * inv_cnt + bias  (3072 waves)
  k_gemm_out<<<768, 128, 0, stream>>>(A, G, invc, bias, out);
  // 6) tuple tail: segment -> float
  k_seg_copy<<<4096, 256, 0, stream>>>(seg, out + (size_t)Bn * NSEG * Fn);
}